// NeuronDecoder_11888469475557
// MI455X (gfx1250) — compile-verified
//
#include <hip/hip_runtime.h>
#include <hip/hip_bf16.h>
#include <math.h>
#include <stdint.h>

// ---------------------------------------------------------------------------
// Shapes: B=2 S=512 E=1024 ; G=4 H=8 Q=16 D=2 V=16 ; N=512
// All matmuls via V_WMMA_F32_16X16X4_F32 (f32 in/out, wave32).
// conn_sub (134 MB stream) moved with the Tensor Data Mover (TDM) into LDS,
// synchronized with s_wait_tensorcnt; B panels / v tiles staged with
// GLOBAL_LOAD_ASYNC_TO_LDS_B32 + s_wait_asynccnt.
// ---------------------------------------------------------------------------

typedef float v2f __attribute__((ext_vector_type(2)));
typedef float v8f __attribute__((ext_vector_type(8)));
typedef unsigned int u32x4 __attribute__((ext_vector_type(4)));
typedef int i32x4 __attribute__((ext_vector_type(4)));
typedef int i32x8 __attribute__((ext_vector_type(8)));

__device__ __forceinline__ v8f wmma_f32_4(v2f a, v2f b, v8f c) {
    return __builtin_amdgcn_wmma_f32_16x16x4_f32(false, a, false, b, (short)0, c,
                                                 false, false);
}

// Async copy one DWORD from global to LDS (ASYNCcnt-tracked).
__device__ __forceinline__ void async_g2l_b32(const float* gp, const float* lp) {
    const uint32_t lo = (uint32_t)(uintptr_t)lp;        // LDS byte offset
    const uint64_t ga = (uint64_t)(uintptr_t)gp;
    asm volatile("global_load_async_to_lds_b32 %0, %1, off"
                 :: "v"(lo), "v"(ga) : "memory");
}
__device__ __forceinline__ void wait_async0() {
    asm volatile("s_wait_asynccnt 0x0" ::: "memory");
}

// TDM: DMA a [16 rows x 512 cols] f32 tile (row stride 512 elems) into LDS.
// D# group0: count=1 | lds_addr | global_addr[56:0] | type=2
// D# group1: data_size=4B; tensor_dim0=512, tensor_dim1=16;
//            tile_dim0=512, tile_dim1=16; tensor_dim0_stride=512.
__device__ __forceinline__ void tdm_load_tile_16x512(const float* gp,
                                                     const float* lp) {
    const uint64_t ga = (uint64_t)(uintptr_t)gp;
    u32x4 g0;
    g0.x = 1u;                                   // count=1 (valid user D#)
    g0.y = (uint32_t)(uintptr_t)lp;              // lds_addr (bytes)
    g0.z = (uint32_t)(ga & 0xFFFFFFFFu);         // global_addr[31:0]
    g0.w = (uint32_t)((ga >> 32) & 0x01FFFFFFu) | (2u << 30);  // [56:32]|type=2
    i32x8 g1;
    g1[0] = 0x00020000;                          // data_size=2 (4B), mask=0
    g1[1] = (int)(512u << 16);                   // tensor_dim0[15:0]=512
    g1[2] = (int)(16u << 16);                    // dim0 hi=0 | tensor_dim1=16
    g1[3] = (int)(512u << 16);                   // dim1 hi=0 | tile_dim0=512
    g1[4] = 16;                                  // tile_dim1=16, tile_dim2=0
    g1[5] = 512;                                 // tensor_dim0_stride=512
    g1[6] = 0;
    g1[7] = 0;
    i32x4 z4 = {0, 0, 0, 0};
#if defined(__clang_major__) && (__clang_major__ >= 23)
    i32x8 z8 = {0, 0, 0, 0, 0, 0, 0, 0};
    __builtin_amdgcn_tensor_load_to_lds(g0, g1, z4, z4, z8, 0);
#else
    __builtin_amdgcn_tensor_load_to_lds(g0, g1, z4, z4, 0);
#endif
}

// ---------------------------------------------------------------------------
// GEMM: C[1024x1024] = A @ Bm + bias. Block = 8 waves sharing one 16-col B
// panel staged in LDS (64KB) via async loads; wave w computes row tile
// mt = (bid>>6)*8 + w at column tile nt = bid & 63. K streamed 4/WMMA.
// ---------------------------------------------------------------------------
__global__ void gemm_wmma_1024(const float* __restrict__ A,
                               const float* __restrict__ Bm,
                               const float* __restrict__ bias,
                               float* __restrict__ C) {
    extern __shared__ float ldsB[];             // [1024][16] = 64KB
    const int tid  = threadIdx.x;
    const int lane = tid & 31;
    const int wave = tid >> 5;
    const int nt   = blockIdx.x & 63;
    const int mt   = (blockIdx.x >> 6) * 8 + wave;
    const int half = lane >> 4;
    const int l    = lane & 15;

    // stage B panel [k=1024][n=16] into LDS asynchronously
    for (int idx = tid; idx < 16384; idx += 256) {
        const int k = idx >> 4, n = idx & 15;
        async_g2l_b32(Bm + (size_t)k * 1024 + nt * 16 + n, ldsB + idx);
    }
    wait_async0();
    __syncthreads();

    const float* arow = A + (size_t)(mt * 16 + l) * 1024 + half * 2;

    v8f acc = {};
#pragma unroll 4
    for (int k0 = 0; k0 < 1024; k0 += 4) {
        v2f a = { arow[k0], arow[k0 + 1] };
        v2f b = { ldsB[(k0 + half * 2) * 16 + l],
                  ldsB[(k0 + half * 2 + 1) * 16 + l] };
        acc = wmma_f32_4(a, b, acc);
    }

    const float bv = bias[nt * 16 + l];
#pragma unroll
    for (int i = 0; i < 8; ++i) {
        C[(size_t)(mt * 16 + half * 8 + i) * 1024 + nt * 16 + l] = acc[i] + bv;
    }
}

// ---------------------------------------------------------------------------
// Attention. grid = B*G*H*8 = 512 blocks, 128 threads (4 waves).
// Dynamic LDS (floats):
//   [0,     8192)   v tile [n=512][vv=16]
//   [8192, 73728)   8 strips of [16][512]: per-wave double-buffered conn/score
//   [73728,73792)   per-wave 1/rowsum
// = 295168 bytes (< 320KB WGP limit).
// Each wave: TDM-load conn tiles for tt=0,1 up-front (tile1 DMA overlaps
// tile0 compute), score WMMAs combine in place, softmax, P@V WMMAs.
// ---------------------------------------------------------------------------
__global__ void attn_wmma(const float* __restrict__ qf,     // [B*S,1024] raw q
                          const float* __restrict__ keys,   // [B,G,N,128]
                          const float* __restrict__ values, // [B,G,N,128]
                          const float* __restrict__ conn,   // [B,G,H,1024,512]
                          float* __restrict__ attn_out) {   // [B,S,1024] cc-order
    extern __shared__ float smem[];
    float* lds_v   = smem;                       // [512][16]
    float* strips  = smem + 8192;                // 8 x [16][512]
    float* lds_r   = smem + 8192 + 65536;        // 4 x [16]

    const int tid  = threadIdx.x;
    const int lane = tid & 31;
    const int wave = tid >> 5;
    const int half = lane >> 4;
    const int l    = lane & 15;

    const int bid    = blockIdx.x;
    const int bgh    = bid >> 3;
    const int rowblk = bid & 7;
    const int b = bgh >> 5;
    const int g = (bgh >> 3) & 3;
    const int h = bgh & 7;
    const int bg = b * 4 + g;

    const size_t conn_base = (((size_t)bg * 8) + h) * 1024 * 512;
    const int row_base = rowblk * 128 + wave * 32;

    // ---- issue TDM loads of both conn tiles into this wave's strips
    tdm_load_tile_16x512(conn + conn_base + (size_t)(row_base +  0) * 512,
                         strips + (wave * 2 + 0) * 8192);
    tdm_load_tile_16x512(conn + conn_base + (size_t)(row_base + 16) * 512,
                         strips + (wave * 2 + 1) * 8192);

    // ---- stage v[n][vv] via async global->LDS copies (L2 resident)
    for (int idx = tid; idx < 512 * 16; idx += 128) {
        const int n = idx >> 4, vv = idx & 15;
        async_g2l_b32(values + ((size_t)bg * 512 + n) * 128 + vv * 8 + h,
                      lds_v + idx);
    }
    wait_async0();
    __syncthreads();

    const float* kp = keys + (size_t)bg * 512 * 128 + h;  // + n*128 + qq*8

    for (int tt = 0; tt < 2; ++tt) {
        const int row0 = row_base + tt * 16;
        float* S = strips + (wave * 2 + tt) * 8192;

        // ---- q A-fragments (K = Q = 16 -> 4 chunks of 4)
        // q[b,g,h,sd,qq] = qf[(b*512+g*128+(sd>>3))*1024 + (sd&7)*128 + qq*8 + h]
        const int m = row0 + l;
        const float* qrow =
            qf + ((size_t)b * 512 + g * 128 + (m >> 3)) * 1024 + (m & 7) * 128 + h;
        v2f afrag[4];
#pragma unroll
        for (int kk = 0; kk < 4; ++kk) {
            const int qq = kk * 4 + half * 2;
            afrag[kk] = { qrow[qq * 8], qrow[qq * 8 + 8] };
        }

        // conn tile for this strip must have landed (TDM completes in order;
        // builtin requires a literal immediate -> two constant call sites)
        if (tt == 0) {
            __builtin_amdgcn_s_wait_tensorcnt(1);
        } else {
            __builtin_amdgcn_s_wait_tensorcnt(0);
        }
        asm volatile("" ::: "memory");

        // ---- scores: 32 n-tiles; combine with conn in place in LDS
        for (int nt = 0; nt < 32; ++nt) {
            const int n0 = nt * 16;
            const float* kpn = kp + (size_t)(n0 + l) * 128;
            v8f acc = {};
#pragma unroll
            for (int kk = 0; kk < 4; ++kk) {
                const int k0 = (kk * 4 + half * 2) * 8;
                v2f bf = { kpn[k0], kpn[k0 + 8] };
                acc = wmma_f32_4(afrag[kk], bf, acc);
            }
#pragma unroll
            for (int i = 0; i < 8; ++i) {
                const int mr = half * 8 + i;
                const float cs = S[mr * 512 + n0 + l];     // conn value (TDM)
                S[mr * 512 + n0 + l] = acc[i] * 0.25f - cs;
            }
        }

        // ---- softmax over n=512: lanes l and l+16 cooperate on row l
        float4* Sr4 = (float4*)(S + l * 512 + half * 256);
        float mx = -3.0e38f;
#pragma unroll 4
        for (int j = 0; j < 64; ++j) {
            float4 t = Sr4[j];
            mx = fmaxf(mx, fmaxf(fmaxf(t.x, t.y), fmaxf(t.z, t.w)));
        }
        mx = fmaxf(mx, __shfl_xor(mx, 16, 32));
        float sum = 0.0f;
#pragma unroll 4
        for (int j = 0; j < 64; ++j) {
            float4 t = Sr4[j];
            t.x = __expf(t.x - mx);
            t.y = __expf(t.y - mx);
            t.z = __expf(t.z - mx);
            t.w = __expf(t.w - mx);
            sum += t.x + t.y + t.z + t.w;
            Sr4[j] = t;
        }
        sum += __shfl_xor(sum, 16, 32);
        if (half == 0) lds_r[wave * 16 + l] = 1.0f / sum;

        // ---- out = P @ v : K = 512 -> 128 WMMAs (A from LDS, B from LDS)
        v8f oacc = {};
#pragma unroll 4
        for (int k0 = 0; k0 < 512; k0 += 4) {
            const int kk = k0 + half * 2;
            v2f a  = { S[l * 512 + kk], S[l * 512 + kk + 1] };
            v2f bf = { lds_v[kk * 16 + l], lds_v[(kk + 1) * 16 + l] };
            oacc = wmma_f32_4(a, bf, oacc);
        }

        // ---- scatter: d = sd>>9, s2 = sd&511, cc = ((g*8+h)*2+d)*16 + vv
#pragma unroll
        for (int i = 0; i < 8; ++i) {
            const int sd = row0 + half * 8 + i;
            const float val = oacc[i] * lds_r[wave * 16 + half * 8 + i];
            const int s2 = sd & 511, d = sd >> 9;
            const int cc = ((g * 8 + h) * 2 + d) * 16 + l;
            attn_out[((size_t)b * 512 + s2) * 1024 + cc] = val;
        }
    }
}

// ---------------------------------------------------------------------------
// Row LayerNorm over 1024 channels. 1024 rows, 256 threads/row.
// ---------------------------------------------------------------------------
__global__ void ln1024(const float* __restrict__ x, const float* __restrict__ gam,
                       const float* __restrict__ bet, float* __restrict__ y) {
    __shared__ float red[16];
    const int row = blockIdx.x;
    const int tid = threadIdx.x;
    const float* xr = x + (size_t)row * 1024;

    float vals[4], s = 0.0f, s2 = 0.0f;
#pragma unroll
    for (int j = 0; j < 4; ++j) {
        const float v = xr[tid + 256 * j];
        vals[j] = v; s += v; s2 += v * v;
    }
#pragma unroll
    for (int off = 16; off >= 1; off >>= 1) {
        s  += __shfl_xor(s,  off, 32);
        s2 += __shfl_xor(s2, off, 32);
    }
    if ((tid & 31) == 0) { red[tid >> 5] = s; red[8 + (tid >> 5)] = s2; }
    __syncthreads();
    float ts = 0.0f, ts2 = 0.0f;
#pragma unroll
    for (int w = 0; w < 8; ++w) { ts += red[w]; ts2 += red[8 + w]; }
    const float mu  = ts * (1.0f / 1024.0f);
    const float var = ts2 * (1.0f / 1024.0f) - mu * mu;
    const float r   = rsqrtf(var + 1e-5f);
#pragma unroll
    for (int j = 0; j < 4; ++j) {
        const int c = tid + 256 * j;
        y[(size_t)row * 1024 + c] = (vals[j] - mu) * r * gam[c] + bet[c];
    }
}

// ---------------------------------------------------------------------------
extern "C" void kernel_launch(void* const* d_in, const int* in_sizes, int n_in,
                              void* d_out, int out_size, void* d_ws, size_t ws_size,
                              hipStream_t stream) {
    const float* emb   = (const float*)d_in[0];
    const float* keys  = (const float*)d_in[1];
    const float* vals  = (const float*)d_in[2];
    const float* conn  = (const float*)d_in[3];
    const float* Wq    = (const float*)d_in[4];
    const float* bq    = (const float*)d_in[5];
    const float* ln_g  = (const float*)d_in[6];
    const float* ln_b  = (const float*)d_in[7];
    const float* Wre   = (const float*)d_in[8];
    const float* bre   = (const float*)d_in[9];
    float* out = (float*)d_out;

    float* q_ws    = (float*)d_ws;               // 4 MB: q / reused for LN out
    float* attn_ws = q_ws + (1u << 20);          // 4 MB: attention output

    // 1) q = emb @ Wq + bq                       (64KB dynamic LDS: B panel)
    gemm_wmma_1024<<<512, 256, 65536, stream>>>(emb, Wq, bq, q_ws);

    // 2) grouped attention; conn_sub streamed by TDM into LDS
    attn_wmma<<<512, 128, 295168, stream>>>(q_ws, keys, vals, conn, attn_ws);

    // 3) LayerNorm (reuses q workspace)
    ln1024<<<1024, 256, 0, stream>>>(attn_ws, ln_g, ln_b, q_ws);

    // 4) final projection: LN @ Wre + bre
    gemm_wmma_1024<<<512, 256, 65536, stream>>>(q_ws, Wre, bre, out);
}